// MCGRUEncoder_17978733101173
// MI455X (gfx1250) — compile-verified
//
#include <hip/hip_runtime.h>

typedef __attribute__((ext_vector_type(2))) float v2f;
typedef __attribute__((ext_vector_type(4))) float v4f;
typedef __attribute__((ext_vector_type(8))) float v8f;

#define Gc   64
#define Dc   16
#define Hc   8
#define Bc   64
#define Tc   512
#define K3   24          // 3*H
#define GD   (Gc * Dc)   // 1024
#define NCH  (Gc * Bc)   // 4096 chains

__device__ __forceinline__ float fast_sigmoid(float x) {
    float e = __expf(-x);
    return __builtin_amdgcn_rcpf(1.0f + e);
}
__device__ __forceinline__ float fast_tanh(float x) {
    float e = __expf(2.0f * x);
    return 1.0f - 2.0f * __builtin_amdgcn_rcpf(e + 1.0f);
}

// ---------------------------------------------------------------------------
// Phase 1: WMMA grouped GEMM.
//   gx[t][g*64+b][k] = sum_d x[b][t][g*16+d] * W_ih[g][k][d] + b_ih[g][k]
// One wave per tile: (g, t, 16 b-rows). K=16 as 4 chained 16x16x4 f32 WMMAs.
// N=24 handled as two OVERLAPPING 16-wide N-tiles (k 0-15 and k 8-23):
// columns 8-15 are computed twice with identical inputs -> identical values,
// so every load and store is unconditional (no exec-mask branching).
// ---------------------------------------------------------------------------
__global__ __launch_bounds__(256) void gru_input_proj_wmma(
    const float* __restrict__ x,      // [B][T][1024]
    const float* __restrict__ W_ih,   // [G][24][16]
    const float* __restrict__ b_ih,   // [G][24]
    float* __restrict__ gx)           // [T][4096][24]
{
    const int lane = threadIdx.x & 31;
    const int wave = threadIdx.x >> 5;
    const int tile = blockIdx.x * 8 + wave;       // 0 .. G*T*4 - 1 (131072)

    const int g    = tile >> 11;                  // / (T*4)
    const int rem  = tile & 2047;
    const int t    = rem >> 2;
    const int b0   = (rem & 3) << 4;

    const int half = lane >> 4;                   // 0: lanes 0-15, 1: 16-31
    const int l16  = lane & 15;

    // ---- A: x rows M = b0+l16 at time t; lane holds K = 4j + 2*half + {0,1}
    v2f a[4];
    {
        const v2f* xp = (const v2f*)(x + ((size_t)(b0 + l16) * Tc + t) * GD
                                       + g * Dc + 2 * half);
        #pragma unroll
        for (int j = 0; j < 4; ++j) a[j] = xp[2 * j];
    }

    // ---- B: W_ih[g][n][d]; tile0: n = l16 (k 0-15), tile1: n = 8+l16 (k 8-23)
    v2f bm0[4], bm1[4];
    {
        const float* wg  = W_ih + (size_t)g * K3 * Dc;
        const v2f*   wr0 = (const v2f*)(wg + (size_t)l16 * Dc + 2 * half);
        const v2f*   wr1 = (const v2f*)(wg + (size_t)(8 + l16) * Dc + 2 * half);
        #pragma unroll
        for (int j = 0; j < 4; ++j) bm0[j] = wr0[2 * j];
        #pragma unroll
        for (int j = 0; j < 4; ++j) bm1[j] = wr1[2 * j];
    }

    // ---- chained WMMA accumulation over K = 0..15
    v8f acc0 = {};
    v8f acc1 = {};
    #pragma unroll
    for (int j = 0; j < 4; ++j) {
        acc0 = __builtin_amdgcn_wmma_f32_16x16x4_f32(
                   false, a[j], false, bm0[j], (short)0, acc0, false, false);
        acc1 = __builtin_amdgcn_wmma_f32_16x16x4_f32(
                   false, a[j], false, bm1[j], (short)0, acc1, false, false);
    }

    // ---- bias + store. C layout: VGPR r -> M = r + 8*half, N = l16.
    const float bias0 = b_ih[g * K3 + l16];       // k = l16
    const float bias1 = b_ih[g * K3 + 8 + l16];   // k = 8 + l16

    const size_t rowBase = (size_t)t * NCH + (size_t)g * 64 + b0;
    #pragma unroll
    for (int r = 0; r < 8; ++r) {
        const int m = r + 8 * half;
        float* orow = gx + (rowBase + m) * K3;
        orow[l16]     = acc0[r] + bias0;          // k 0..15
        orow[8 + l16] = acc1[r] + bias1;          // k 8..23 (8-15 double-write)
    }
}

// ---------------------------------------------------------------------------
// Phase 2: serial GRU recurrence, one lane per (b,g) chain (4096 chains,
// 128 wave32 waves). W_hh/b_hh hoisted into registers (wave-uniform values);
// gx loads software-pipelined one step ahead (ping-pong, unroll-by-2).
// ---------------------------------------------------------------------------
__global__ __launch_bounds__(256) void gru_recurrence(
    const float* __restrict__ gx,     // [T][4096][24]
    const float* __restrict__ W_hh,   // [G][24][8]
    const float* __restrict__ b_hh,   // [G][24]
    float* __restrict__ out)          // [B][T][G][8]
{
    __shared__ float sW[4 * K3 * Hc];   // 4 groups per block (staging only)
    __shared__ float sB[4 * K3];

    const int tid   = threadIdx.x;
    const int gBase = blockIdx.x * 4;

    for (int i = tid; i < 4 * K3 * Hc; i += 256)
        sW[i] = W_hh[(size_t)gBase * K3 * Hc + i];
    for (int i = tid; i < 4 * K3; i += 256)
        sB[i] = b_hh[(size_t)gBase * K3 + i];
    __syncthreads();

    const int c  = blockIdx.x * 256 + tid;  // chain id: g = c>>6, b = c&63
    const int b  = c & 63;
    const int gl = tid >> 6;                // wave-uniform local group id
    const int g  = gBase + gl;

    // hoist recurrent weights + bias into registers (192 + 24 VGPRs)
    float w[K3][Hc];
    float bb[K3];
    {
        const v4f* sWv = (const v4f*)(sW + gl * K3 * Hc);
        #pragma unroll
        for (int k = 0; k < K3; ++k) {
            v4f w0 = sWv[2 * k];
            v4f w1 = sWv[2 * k + 1];
            w[k][0] = w0.x; w[k][1] = w0.y; w[k][2] = w0.z; w[k][3] = w0.w;
            w[k][4] = w1.x; w[k][5] = w1.y; w[k][6] = w1.z; w[k][7] = w1.w;
            bb[k]   = sB[gl * K3 + k];
        }
    }

    float* outBase  = out + (size_t)b * Tc * (Gc * Hc) + (size_t)g * Hc;
    const float* gc = gx + (size_t)c * K3;

    float h[Hc];
    #pragma unroll
    for (int i = 0; i < Hc; ++i) h[i] = 0.0f;

    auto loadgx = [&](v4f buf[6], int t) {
        const v4f* gp = (const v4f*)(gc + (size_t)t * (NCH * K3));
        #pragma unroll
        for (int i = 0; i < 6; ++i) buf[i] = gp[i];
    };

    auto step = [&](const v4f buf[6], int t) {
        float gxa[K3];
        #pragma unroll
        for (int i = 0; i < 6; ++i) {
            gxa[4 * i + 0] = buf[i].x; gxa[4 * i + 1] = buf[i].y;
            gxa[4 * i + 2] = buf[i].z; gxa[4 * i + 3] = buf[i].w;
        }

        // gh = W_hh @ h + b_hh  (pure register FMAs)
        float gh[K3];
        #pragma unroll
        for (int k = 0; k < K3; ++k) {
            float acc = bb[k];
            #pragma unroll
            for (int j = 0; j < Hc; ++j) acc += w[k][j] * h[j];
            gh[k] = acc;
        }

        // gates (PyTorch order r, z, n) + state update
        #pragma unroll
        for (int i = 0; i < Hc; ++i) {
            float r = fast_sigmoid(gxa[i]      + gh[i]);
            float z = fast_sigmoid(gxa[8 + i]  + gh[8 + i]);
            float n = fast_tanh(gxa[16 + i] + r * gh[16 + i]);
            h[i] = (1.0f - z) * n + z * h[i];
        }

        v4f o0 = {h[0], h[1], h[2], h[3]};
        v4f o1 = {h[4], h[5], h[6], h[7]};
        v4f* op = (v4f*)(outBase + (size_t)t * (Gc * Hc));
        op[0] = o0;
        op[1] = o1;
    };

    // ping-pong double buffer, unroll-by-2 (no dynamic buffer indexing)
    v4f bufA[6], bufB[6];
    loadgx(bufA, 0);
    for (int t = 0; t < Tc; t += 2) {
        loadgx(bufB, t + 1);          // prefetch t+1 before computing t
        step(bufA, t);
        if (t + 2 < Tc) loadgx(bufA, t + 2);
        step(bufB, t + 1);
    }
}

// ---------------------------------------------------------------------------
extern "C" void kernel_launch(void* const* d_in, const int* in_sizes, int n_in,
                              void* d_out, int out_size, void* d_ws, size_t ws_size,
                              hipStream_t stream) {
    const float* x    = (const float*)d_in[0];
    const float* W_ih = (const float*)d_in[1];
    const float* W_hh = (const float*)d_in[2];
    const float* b_ih = (const float*)d_in[3];
    const float* b_hh = (const float*)d_in[4];
    float* out = (float*)d_out;
    float* gx  = (float*)d_ws;   // needs T*4096*24*4 = 192 MB scratch

    // Phase 1: G*T*(B/16) = 131072 wave-tiles, 8 waves per 256-thread block
    gru_input_proj_wmma<<<dim3(131072 / 8), dim3(256), 0, stream>>>(x, W_ih, b_ih, gx);
    // Phase 2: 4096 chains -> 16 blocks of 256 threads (128 wave32 waves)
    gru_recurrence<<<dim3(NCH / 256), dim3(256), 0, stream>>>(gx, W_hh, b_hh, out);
}